// spectralAgg_30116310680041
// MI455X (gfx1250) — compile-verified
//
#include <hip/hip_runtime.h>
#include <hip/hip_bf16.h>
#include <stdint.h>

// DANet-style per-patch channel attention for x:(8,64,512,512) fp32, COUNT=8.
// 512 independent problems: p (64x4096) -> energy = p p^T -> softmax(rowmin trick)
// -> out = (1+gamma) * att * p. One 256-thread workgroup per problem.
// fp32 WMMA (v_wmma_f32_16x16x4_f32) + Tensor Data Mover staging (double buffered).

typedef __attribute__((ext_vector_type(2))) float v2f;
typedef __attribute__((ext_vector_type(8))) float v8f;
typedef __attribute__((ext_vector_type(4))) uint32_t u32x4;
typedef __attribute__((ext_vector_type(8))) uint32_t u32x8;

#define HW      262144   // 512*512
#define CHW     16777216 // 64*512*512
#define CHUNK   64       // i-elements per staged chunk = one patch row
#define NCHUNK  64       // 4096 / 64
#define PSTRIDE 68       // LDS floats per channel row (64 + 4 pad, via TDM pad)
#define BUFBYTES (64 * PSTRIDE * 4)   // 17408 bytes per chunk buffer

// Issue one TDM load: an 8-channel x 64-float 2D tile -> LDS, with hardware
// padding of 4 dwords after every 64 dwords (producing the 68-float LDS rows).
__device__ __forceinline__ void tdm_issue(uint64_t gaddr, uint32_t lds_byte) {
    u32x4 g0;
    g0[0] = 1u;                                     // count=1, user descriptor
    g0[1] = lds_byte;                               // lds_addr (bytes)
    g0[2] = (uint32_t)gaddr;                        // global_addr[31:0]
    g0[3] = (uint32_t)(gaddr >> 32) | 0x80000000u;  // global_addr[56:32] | type=2
    u32x8 g1;
    g1[0] = 0x07520000u;   // data_size=4B, pad_enable, pad_interval=64dw, pad_amount=4dw
    g1[1] = 64u << 16;     // tensor_dim0 = 64 (x extent)
    g1[2] = 8u << 16;      // tensor_dim1 = 8 (channel slice extent)
    g1[3] = 64u << 16;     // tile_dim0 = 64
    g1[4] = 8u;            // tile_dim1 = 8, tile_dim2 = 0 (2D)
    g1[5] = 512u;          // tensor_dim0_stride = 512 elements (image row pitch)
    g1[6] = 0u;
    g1[7] = 0u;
    asm volatile("tensor_load_to_lds %0, %1"
                 :
                 : "s"(g0), "s"(g1)
                 : "memory");
}

__global__ void __launch_bounds__(256)
danet_patch_attn_kernel(const float* __restrict__ x,
                        const float* __restrict__ gamma,
                        float* __restrict__ out) {
    __shared__ __align__(16) float pl[2][64][PSTRIDE]; // double-buffered p chunk
    __shared__ float el[64][PSTRIDE];                  // energy -> attention

    const int tid  = threadIdx.x;
    const int lane = tid & 31;
    const int wave = tid >> 5;
    const int h    = lane >> 4;   // half-wave: selects K pair / M+8
    const int m    = lane & 15;   // row/col within tile

    const int prob  = blockIdx.x;          // 0..511
    const int b     = prob >> 6;
    const int patch = prob & 63;
    const int pr    = patch >> 3;
    const int pc    = patch & 7;
    const unsigned base = (unsigned)b * CHW + (unsigned)(pr * 64 * 512 + pc * 64);
    const float* src = x + base;
    float*       dst = out + base;

    // Per-wave TDM slice: channels [wave*8, wave*8+8).
    const uint64_t gbase   = (uint64_t)(uintptr_t)src + (uint64_t)wave * (8ull * HW * 4ull);
    const uint32_t ldsbase = (uint32_t)(uintptr_t)(&pl[0][0][0]) + (uint32_t)wave * (8u * PSTRIDE * 4u);

    // ---------------- Phase 1: energy = p p^T (64x64) ----------------
    // 16 tiles (4x4 grid of 16x16); wave w owns tiles (cm, cn0) and (cm, cn1).
    const int cm  = wave >> 1;
    const int cn0 = (wave & 1) * 2;
    const int cn1 = cn0 + 1;

    v8f acc0 = {};
    v8f acc1 = {};

    tdm_issue(gbase, ldsbase);   // chunk 0 -> buffer 0
    for (int cc = 0; cc < NCHUNK; ++cc) {
        const int buf = cc & 1;
        __builtin_amdgcn_s_wait_tensorcnt(0);   // own slice of chunk cc landed
        __syncthreads();                        // all slices landed; prev reads done
        if (cc + 1 < NCHUNK)
            tdm_issue(gbase + (uint64_t)(cc + 1) * 2048ull,
                      ldsbase + (uint32_t)(buf ^ 1) * BUFBYTES);

        const float* ar  = &pl[buf][cm  * 16 + m][0];
        const float* b0r = &pl[buf][cn0 * 16 + m][0];
        const float* b1r = &pl[buf][cn1 * 16 + m][0];
#pragma unroll
        for (int ks = 0; ks < CHUNK / 4; ++ks) {
            const int k = ks * 4 + 2 * h;   // lanes 0-15: K=0,1  lanes 16-31: K=2,3
            v2f a, b0, b1;
            a.x  = ar[k];   a.y  = ar[k + 1];
            b0.x = b0r[k];  b0.y = b0r[k + 1];
            b1.x = b1r[k];  b1.y = b1r[k + 1];
            acc0 = __builtin_amdgcn_wmma_f32_16x16x4_f32(
                false, a, false, b0, (short)0, acc0, false, false);
            acc1 = __builtin_amdgcn_wmma_f32_16x16x4_f32(
                false, a, false, b1, (short)0, acc1, false, false);
        }
    }
    __syncthreads();

    // Spill energy tiles to LDS. C/D layout: VGPR v -> M = v + 8*h, N = m.
#pragma unroll
    for (int v = 0; v < 8; ++v) {
        el[cm * 16 + v + 8 * h][cn0 * 16 + m] = acc0[v];
        el[cm * 16 + v + 8 * h][cn1 * 16 + m] = acc1[v];
    }
    __syncthreads();

    // ---------------- Phase 2: softmax(rowmax - e) == exp(rowmin - e)/sum ----
    if (tid < 64) {
        float mn = el[tid][0];
        for (int d = 1; d < 64; ++d) mn = fminf(mn, el[tid][d]);
        float s = 0.0f;
        for (int d = 0; d < 64; ++d) s += __expf(mn - el[tid][d]);
        const float inv = 1.0f / s;
        for (int d = 0; d < 64; ++d) el[tid][d] = __expf(mn - el[tid][d]) * inv;
    }
    __syncthreads();

    // ---------------- Phase 3: out = (1+gamma) * att * p ----------------
    const float scale = 1.0f + gamma[0];
    const int   m3    = wave >> 1;         // M block (output channel block)
    const int   nb    = (wave & 1) * 2;    // first of 2 owned N tiles per chunk

    // Hoist attention A fragments: K = 64 -> 16 k-steps, fixed for all chunks.
    v2f aw[16];
#pragma unroll
    for (int ks = 0; ks < 16; ++ks) {
        const int k = ks * 4 + 2 * h;
        aw[ks].x = el[m3 * 16 + m][k];
        aw[ks].y = el[m3 * 16 + m][k + 1];
    }

    tdm_issue(gbase, ldsbase);   // restart pipeline: chunk 0 -> buffer 0
    for (int cc = 0; cc < NCHUNK; ++cc) {
        const int buf = cc & 1;
        __builtin_amdgcn_s_wait_tensorcnt(0);
        __syncthreads();
        if (cc + 1 < NCHUNK)
            tdm_issue(gbase + (uint64_t)(cc + 1) * 2048ull,
                      ldsbase + (uint32_t)(buf ^ 1) * BUFBYTES);

#pragma unroll
        for (int nt = 0; nt < 2; ++nt) {
            const int i0 = (nb + nt) * 16;
            v8f acc = {};
#pragma unroll
            for (int ks = 0; ks < 16; ++ks) {
                const int d0 = ks * 4 + 2 * h;  // B rows are channel index d
                v2f bb;
                bb.x = pl[buf][d0][i0 + m];
                bb.y = pl[buf][d0 + 1][i0 + m];
                acc = __builtin_amdgcn_wmma_f32_16x16x4_f32(
                    false, aw[ks], false, bb, (short)0, acc, false, false);
            }
            // Write tile: patch row y = cc, columns i0 + m, rows m3*16 + v + 8h.
            const int xx = i0 + m;
#pragma unroll
            for (int v = 0; v < 8; ++v) {
                const int c = m3 * 16 + v + 8 * h;
                dst[c * HW + cc * 512 + xx] = scale * acc[v];
            }
        }
    }
}

extern "C" void kernel_launch(void* const* d_in, const int* in_sizes, int n_in,
                              void* d_out, int out_size, void* d_ws, size_t ws_size,
                              hipStream_t stream) {
    const float* x     = (const float*)d_in[0];
    const float* gamma = (const float*)d_in[1];
    float*       out   = (float*)d_out;
    // 512 problems = 8 batches * 64 patches; one 256-thread workgroup each.
    danet_patch_attn_kernel<<<dim3(512), dim3(256), 0, stream>>>(x, gamma, out);
}